// CodeBook_45758581572167
// MI455X (gfx1250) — compile-verified
//
#include <hip/hip_runtime.h>
#include <hip/hip_bf16.h>

typedef __attribute__((ext_vector_type(2))) float v2f;
typedef __attribute__((ext_vector_type(8))) float v8f;

#define K_ROWS   8192
#define DIM      64
#define N_IN     2097152      // 8*64*64*64
#define N_IDX    32768        // 8*64*64
#define NBLK2    512

// ---------------- workspace layout ----------------
// ws[0..7]   : u64 packed (norm_bits<<32 | row)  -- argmin key
// ws+16      : float partials[NBLK2]             -- per-block loss partial sums
// required ws_size >= 16 + NBLK2*4 = 2064 bytes

__global__ void vq_init(unsigned long long* key) {
    *key = ~0ull;
}

// One wave (32 lanes) per 16 book rows. Gram diagonal via V_WMMA_F32_16X16X4_F32.
__global__ __launch_bounds__(32) void vq_norm_argmin(const float* __restrict__ book,
                                                     unsigned long long* __restrict__ key) {
    const int lane = threadIdx.x;          // 0..31, all active -> EXEC all ones
    const int base = blockIdx.x * 16;      // 512 blocks cover 8192 rows
    const int r    = lane & 15;
    const int hk   = (lane >> 4) * 2;
    const float* rowp = book + (base + r) * DIM + hk;

    v8f acc = {};
#pragma unroll
    for (int d0 = 0; d0 < DIM; d0 += 4) {
        // A(16x4) lane layout == B(4x16) layout for chunk * chunk^T : reuse one float2
        v2f a;
        a.x = rowp[d0 + 0];
        a.y = rowp[d0 + 1];
        // D = A*B + C ; D[i][i] accumulates ||book[base+i]||^2
        acc = __builtin_amdgcn_wmma_f32_16x16x4_f32(
            /*neg_a=*/false, a, /*neg_b=*/false, a,
            /*c_mod=*/(short)0, acc, /*reuse_a=*/false, /*reuse_b=*/false);
    }

    // Diagonal extraction: lanes 0..7 -> comp=lane (rows base+0..7),
    //                      lanes 24..31 -> comp=lane-24 (rows base+8..15).
    // Branch-free v_cndmask select chain (avoids dynamic-index scratch spill).
    const int comp = (lane < 8) ? lane : (lane - 24);
    float norm = acc[0];
#pragma unroll
    for (int i = 1; i < 8; ++i)
        norm = (comp == i) ? acc[i] : norm;

    unsigned long long k = ~0ull;
    if (lane < 8) {
        k = ((unsigned long long)__float_as_uint(norm) << 32) |
            (unsigned)(base + lane);
    } else if (lane >= 24) {
        k = ((unsigned long long)__float_as_uint(norm) << 32) |
            (unsigned)(base + lane - 16);
    }
    // wave32 min-reduce (packed key: smaller norm wins, ties -> smaller row index)
#pragma unroll
    for (int off = 16; off > 0; off >>= 1) {
        unsigned long long o = __shfl_xor(k, off, 32);
        k = (o < k) ? o : k;
    }
    if (lane == 0) atomicMin(key, k);
}

// Streaming kernel: z_q output, idx output, per-block loss partials.
__global__ __launch_bounds__(256) void vq_fused(const float* __restrict__ in,
                                                const float* __restrict__ book,
                                                const unsigned long long* __restrict__ key,
                                                float* __restrict__ zq_out,
                                                float* __restrict__ idx_out,
                                                float* __restrict__ partials) {
    __shared__ float sq[DIM];
    __shared__ float red[256];

    const unsigned kstar = (unsigned)(*key & 0xffffffffull);
    if (threadIdx.x < DIM) sq[threadIdx.x] = book[(size_t)kstar * DIM + threadIdx.x];
    __syncthreads();

    const int tid = blockIdx.x * 256 + threadIdx.x;
    const int nth = NBLK2 * 256;                 // 131072 threads, 4 float4 each

    const float4* in4  = (const float4*)in;
    float4*       out4 = (float4*)zq_out;

    float accum = 0.0f;
    for (int t = tid; t < N_IN / 4; t += nth) {
        float4 x = in4[t];
        int j = t * 4;                            // flat BCHW index; w = j & 63
        int wb = j & 63;                          // aligned: wb in {0,4,...,60}
        float d0 = sq[wb + 0] - x.x;
        float d1 = sq[wb + 1] - x.y;
        float d2 = sq[wb + 2] - x.z;
        float d3 = sq[wb + 3] - x.w;
        accum += d0 * d0 + d1 * d1 + d2 * d2 + d3 * d3;
        // output layout [B,W,C,H]: value depends only on w = (flat>>12)&63,
        // constant across the 4 packed elements
        float qo = sq[(j >> 12) & 63];
        float4 qv = { qo, qo, qo, qo };
        out4[t] = qv;
    }

    if (tid < N_IDX) idx_out[tid] = (float)kstar;

    // deterministic intra-block reduction
    red[threadIdx.x] = accum;
    __syncthreads();
#pragma unroll
    for (int s = 128; s > 0; s >>= 1) {
        if (threadIdx.x < s) red[threadIdx.x] += red[threadIdx.x + s];
        __syncthreads();
    }
    if (threadIdx.x == 0) partials[blockIdx.x] = red[0];
}

// Deterministic final reduce (fixed order) + loss scale.
__global__ __launch_bounds__(512) void vq_finalize(const float* __restrict__ partials,
                                                   float* __restrict__ loss_out) {
    __shared__ float red[NBLK2];
    red[threadIdx.x] = partials[threadIdx.x];
    __syncthreads();
#pragma unroll
    for (int s = NBLK2 / 2; s > 0; s >>= 1) {
        if (threadIdx.x < s) red[threadIdx.x] += red[threadIdx.x + s];
        __syncthreads();
    }
    if (threadIdx.x == 0)
        *loss_out = 1.25f * red[0] / (float)N_IN;   // mean + BETA*mean, BETA=0.25
}

extern "C" void kernel_launch(void* const* d_in, const int* in_sizes, int n_in,
                              void* d_out, int out_size, void* d_ws, size_t ws_size,
                              hipStream_t stream) {
    const float* in   = (const float*)d_in[0];   // [8,64,64,64] f32
    const float* book = (const float*)d_in[1];   // [8192,64] f32
    float* out = (float*)d_out;                  // [z_q | idx | loss] = 2097152+32768+1

    unsigned long long* key = (unsigned long long*)d_ws;
    float* partials = (float*)((char*)d_ws + 16);

    vq_init<<<1, 1, 0, stream>>>(key);
    vq_norm_argmin<<<K_ROWS / 16, 32, 0, stream>>>(book, key);
    vq_fused<<<NBLK2, 256, 0, stream>>>(in, book, key,
                                        out, out + N_IN, partials);
    vq_finalize<<<1, NBLK2, 0, stream>>>(partials, out + N_IN + N_IDX);
}